// CompressDCT_9543417331869
// MI455X (gfx1250) — compile-verified
//
#include <hip/hip_runtime.h>

// CDNA5 / gfx1250 8x8 block DCT (JPEG-style encoder/decoder) using
// V_WMMA_F32_16X16X4_F32. Memory-bound (4 FLOP/byte, ~23us traffic floor at
// 23.3 TB/s): each 16-wave workgroup processes one 16x256 row strip; each
// wave handles one 16x16 tile (= 2x2 group of 8x8 DCT blocks) with 8 chained
// f32 WMMAs against the block-diagonal coefficient matrix diag(D, D).

typedef __attribute__((ext_vector_type(2))) float v2f;
typedef __attribute__((ext_vector_type(8))) float v8f;

#define TILE_W 256
#define WAVES  16

__global__ __launch_bounds__(WAVES * 32)
void dct8x8_wmma(const float* __restrict__ x,
                 const float* __restrict__ qt,
                 const int*   __restrict__ is_enc_p,
                 float*       __restrict__ out)
{
    // 64-entry DCT coefficient table + quantization table (built once per WG;
    // Qs holds 1/q for the encoder so the hot path is a multiply, q for decoder)
    __shared__ float Dmat[8][8];
    __shared__ float Qs[64];
    // per-wave transpose buffer for stage-1 result (stride 18: conflict-free,
    // 8-byte aligned pairs for the stage-2 A-operand reload)
    __shared__ float Tbuf[WAVES][16][18];

    const int tid  = threadIdx.x;
    const int wave = tid >> 5;
    const int lane = tid & 31;
    const int hi   = lane >> 4;   // lane half: selects K/M offset per WMMA layout
    const int r15  = lane & 15;
    const int is_enc = *is_enc_p;

    if (tid < 64) {
        const int i = tid >> 3, j = tid & 7;
        const float s = (i == 0) ? 0.35355339059327373f : 0.5f;
        Dmat[i][j] = s * cosf(3.14159265358979323846f * (float)((2 * j + 1) * i) / 16.0f);
    } else if (tid < 128) {
        const int i = tid - 64;
        const float qv = qt[i];
        Qs[i] = is_enc ? (1.0f / qv) : qv;   // 64 divisions amortized over 4096 outputs
    }
    __syncthreads();

    // Per-lane constant operand registers.
    // Encoder:  stage-1 A = Dblk,  stage-2 B = Dblk^T  -> both = Dblk(r15, k)
    // Decoder:  stage-1 A = Dblk^T, stage-2 B = Dblk   -> both = Dblk(k, r15)
    // where Dblk = diag(D, D) (block-diagonal 16x16).
    float coef[8];
    float qreg[8];
#pragma unroll
    for (int jj = 0; jj < 8; ++jj) {
        const int k  = 4 * (jj >> 1) + 2 * hi + (jj & 1);
        const int rr = is_enc ? r15 : k;
        const int cc = is_enc ? k : r15;
        coef[jj] = ((rr ^ cc) & 8) ? 0.0f : Dmat[rr & 7][cc & 7];
        // Encoder: qreg[r] = 1/q[r][lane&7] for output row r (row mod 8 == vgpr idx).
        // Decoder: qreg[jj] = q[k&7][lane&7] for pre-scaling input element (k, r15).
        // LDS read pattern: 8 distinct banks, 4-way same-address broadcast each.
        qreg[jj] = is_enc ? Qs[jj * 8 + (lane & 7)]
                          : Qs[(k & 7) * 8 + (lane & 7)];
    }

    const size_t strip = blockIdx.x;
    const float* src = x   + strip * (size_t)(16 * TILE_W);
    float*       dst = out + strip * (size_t)(16 * TILE_W);
    const int col = (wave << 4) + r15;   // column of this lane within the strip

    // ---- Stage 1: T = C1 * tile  (tile loaded directly in B-operand layout) ----
    // B(k, n): vgpr v, lanes 0-15 hold K = 4c+v, lanes 16-31 hold K = 4c+2+v.
    // Uniform branch on is_enc keeps the decoder pre-scale out of the encoder path.
    v8f t = {};
    if (is_enc) {
#pragma unroll
        for (int c = 0; c < 4; ++c) {
            const int k0 = 4 * c + 2 * hi;
            v2f a; a.x = coef[2 * c]; a.y = coef[2 * c + 1];
            v2f b;
            b.x = src[(size_t)(k0)     * TILE_W + col];
            b.y = src[(size_t)(k0 + 1) * TILE_W + col];
            t = __builtin_amdgcn_wmma_f32_16x16x4_f32(
                    false, a, false, b, (short)0, t, false, false);
        }
    } else {
#pragma unroll
        for (int c = 0; c < 4; ++c) {
            const int k0 = 4 * c + 2 * hi;
            v2f a; a.x = coef[2 * c]; a.y = coef[2 * c + 1];
            v2f b;
            b.x = src[(size_t)(k0)     * TILE_W + col] * qreg[2 * c];
            b.y = src[(size_t)(k0 + 1) * TILE_W + col] * qreg[2 * c + 1];
            t = __builtin_amdgcn_wmma_f32_16x16x4_f32(
                    false, a, false, b, (short)0, t, false, false);
        }
    }

    // ---- D-layout -> A-layout via per-wave LDS tile ----
    // D layout: vgpr r, lanes 0-15: (M=r, N=lane); lanes 16-31: (M=r+8, N=lane-16)
#pragma unroll
    for (int r = 0; r < 8; ++r)
        Tbuf[wave][r + 8 * hi][r15] = t[r];

    // In-wave store->load ordering (cross-lane through LDS, single PC per wave).
    asm volatile("s_wait_dscnt 0" ::: "memory");

    // ---- Stage 2: X = T * C2 ----
    // A(m, k): lanes 0-15 hold (m=r15, k=4c+v); lanes 16-31 hold (m=r15, k=4c+2+v)
    v8f X = {};
#pragma unroll
    for (int c = 0; c < 4; ++c) {
        const int k0 = 4 * c + 2 * hi;
        v2f a; a.x = Tbuf[wave][r15][k0]; a.y = Tbuf[wave][r15][k0 + 1];
        v2f b; b.x = coef[2 * c];         b.y = coef[2 * c + 1];
        X = __builtin_amdgcn_wmma_f32_16x16x4_f32(
                false, a, false, b, (short)0, X, false, false);
    }

    // ---- Quantize / clip / store (row-contiguous, coalesced) ----
    if (is_enc) {
#pragma unroll
        for (int r = 0; r < 8; ++r) {
            float v = rintf(X[r] * qreg[r]);         // round-half-even (RNE mode)
            v = fminf(fmaxf(v, -128.0f), 127.0f);    // clip to [-2^7, 2^7-1]
            dst[(size_t)(r + 8 * hi) * TILE_W + col] = v;
        }
    } else {
#pragma unroll
        for (int r = 0; r < 8; ++r)
            dst[(size_t)(r + 8 * hi) * TILE_W + col] = X[r];
    }
}

extern "C" void kernel_launch(void* const* d_in, const int* in_sizes, int n_in,
                              void* d_out, int out_size, void* d_ws, size_t ws_size,
                              hipStream_t stream) {
    (void)in_sizes; (void)n_in; (void)d_ws; (void)ws_size;
    const float* x      = (const float*)d_in[0];
    const float* qt     = (const float*)d_in[1];
    const int*   is_enc = (const int*)d_in[2];
    float*       out    = (float*)d_out;

    // out_size = N*C*H*W = 67,108,864 ; one WG per 16x256 strip
    const int strips = out_size / (16 * TILE_W);   // 16384
    dct8x8_wmma<<<dim3(strips), dim3(WAVES * 32), 0, stream>>>(x, qt, is_enc, out);
}